// CAPooLayer_79207786873547
// MI455X (gfx1250) — compile-verified
//
#include <hip/hip_runtime.h>
#include <hip/hip_bf16.h>

typedef __attribute__((ext_vector_type(2))) float v2f;
typedef __attribute__((ext_vector_type(8))) float v8f;

#define N_GRAPHS 64
#define SEG      16384
#define N_TOTAL  (N_GRAPHS * SEG)      // 1,048,576
#define KKEEP    8192                  // ceil(0.5 * SEG)
#define N_KEPT   (N_GRAPHS * KKEEP)    // 524,288
#define F_IN     64
#define N_EDGES  (4 * N_TOTAL)         // 4,194,304
#define RO_SLICES 8                    // readout parallelism per graph
#define SLICE_ROWS (KKEEP / RO_SLICES) // 1024

static __device__ __forceinline__ v8f wmma_f32_4(v2f a, v2f b, v8f c) {
  // D = A(16x4,f32) x B(4x16,f32) + C(16x16,f32), full f32 precision
  return __builtin_amdgcn_wmma_f32_16x16x4_f32(false, a, false, b, (short)0, c,
                                               false, false);
}

// ---------------------------------------------------------------------------
// K1: zp[i] = dot(x[i,:], w)  via chained V_WMMA_F32_16X16X4_F32.
// One wave handles 16 rows; 16 WMMAs walk K=64 in chunks of 4.
// B is w broadcast across all 16 columns, so every D column holds the score.
// A layout (ISA 7.12.2): lane&15 = M; lane>>4 selects K pair {0,1} vs {2,3}.
// ---------------------------------------------------------------------------
__global__ __launch_bounds__(256) void score_kernel(
    const float* __restrict__ x, const float* __restrict__ w,
    float* __restrict__ zp) {
  const int wave = blockIdx.x * (blockDim.x >> 5) + (threadIdx.x >> 5);
  const int lane = threadIdx.x & 31;
  const int m    = lane & 15;
  const int half = lane >> 4;
  const int row0 = wave * 16;
  const float* rowp = x + (size_t)(row0 + m) * F_IN + half * 2;
  v8f c = {};
#pragma unroll
  for (int kc = 0; kc < 16; ++kc) {
    v2f a; a.x = rowp[kc * 4 + 0];            a.y = rowp[kc * 4 + 1];
    v2f b; b.x = w[kc * 4 + half * 2 + 0];    b.y = w[kc * 4 + half * 2 + 1];
    c = wmma_f32_4(a, b, c);
  }
  // D VGPR j: lanes0-15 = D[j][n], lanes16-31 = D[8+j][n]; all n equal.
  if (m == 0) {
#pragma unroll
    for (int j = 0; j < 8; ++j) zp[row0 + half * 8 + j] = c[j];
  }
}

// ---------------------------------------------------------------------------
// K2: per-graph top-KKEEP via 4x8-bit MSB-first LDS-histogram radix select,
// then deterministic ordered compaction (block prefix scan, wave32 shuffles).
// One 1024-thread block (32 wave32) per graph; 16 contiguous keys per thread.
// zp (4 MB) is L2-resident (192 MB L2) right after K1, so 64 blocks suffice.
// ---------------------------------------------------------------------------
static __device__ __forceinline__ unsigned ordkey(float f) {
  unsigned v = __float_as_uint(f);
  return (v & 0x80000000u) ? ~v : (v | 0x80000000u);  // monotone: big f -> big u
}

__global__ __launch_bounds__(1024) void topk_kernel(
    const float* __restrict__ zp, int* __restrict__ idx_out) {
  const int b = blockIdx.x;
  const int t = threadIdx.x;
  const int lane = t & 31, wid = t >> 5;

  __shared__ unsigned hist[256];
  __shared__ unsigned wsum[32];
  __shared__ unsigned s_prefix;
  __shared__ int s_krem;

  // 16 contiguous keys per thread (index order preserved inside a thread).
  const float4* z4 = (const float4*)(zp + (size_t)b * SEG + t * 16);
  unsigned keys[16];
#pragma unroll
  for (int c = 0; c < 4; ++c) {
    float4 v = z4[c];
    keys[c * 4 + 0] = ordkey(v.x);
    keys[c * 4 + 1] = ordkey(v.y);
    keys[c * 4 + 2] = ordkey(v.z);
    keys[c * 4 + 3] = ordkey(v.w);
  }

  if (t == 0) { s_prefix = 0u; s_krem = KKEEP; }
  __syncthreads();

  for (int pass = 0; pass < 4; ++pass) {
    const int sh = 24 - pass * 8;
    if (t < 256) hist[t] = 0u;
    __syncthreads();
    const unsigned prefix = s_prefix;
    const unsigned him = (pass == 0) ? 0u : (0xFFFFFFFFu << (sh + 8));
#pragma unroll
    for (int i = 0; i < 16; ++i) {
      const unsigned u = keys[i];
      if ((u & him) == prefix) atomicAdd(&hist[(u >> sh) & 255u], 1u);
    }
    __syncthreads();
    if (t == 0) {
      int krem = s_krem;
      unsigned cum = 0; int bsel = 0;
      for (int bin = 255; bin >= 0; --bin) {
        const unsigned c = hist[bin];
        if (cum + c >= (unsigned)krem) { bsel = bin; break; }
        cum += c;
      }
      s_prefix = prefix | ((unsigned)bsel << sh);
      s_krem = krem - (int)cum;
    }
    __syncthreads();
  }

  const unsigned T = s_prefix;      // KKEEP-th largest key
  const int neq_keep = s_krem;      // # of ==T to keep (index order)
  const int G = KKEEP - neq_keep;   // # strictly greater than T

  // per-thread class counts
  int cg = 0, ce = 0;
#pragma unroll
  for (int i = 0; i < 16; ++i) { cg += (keys[i] > T); ce += (keys[i] == T); }

  // ordered block-exclusive scan of packed (cg<<16 | ce)
  unsigned v = ((unsigned)cg << 16) | (unsigned)ce;
  unsigned inc = v;
#pragma unroll
  for (int d = 1; d < 32; d <<= 1) {
    unsigned n = __shfl_up(inc, d, 32);
    if (lane >= d) inc += n;
  }
  if (lane == 31) wsum[wid] = inc;
  __syncthreads();
  if (wid == 0) {
    unsigned wv = wsum[lane];
    unsigned winc = wv;
#pragma unroll
    for (int d = 1; d < 32; d <<= 1) {
      unsigned n = __shfl_up(winc, d, 32);
      if (lane >= d) winc += n;
    }
    wsum[lane] = winc - wv;  // exclusive wave base
  }
  __syncthreads();
  const unsigned exc = inc - v + wsum[wid];
  int pg = (int)(exc >> 16);
  int pe = (int)(exc & 0xFFFFu);

  int* out = idx_out + (size_t)b * KKEEP;
#pragma unroll
  for (int i = 0; i < 16; ++i) {
    const unsigned u = keys[i];
    const int cell = b * SEG + t * 16 + i;
    if (u > T) { out[pg] = cell; ++pg; }
    else if (u == T) { if (pe < neq_keep) out[G + pe] = cell; ++pe; }
  }
}

// ---------------------------------------------------------------------------
// K3: x_p[j,:] = x[idx[j],:] * relu(zp[idx[j]]);  new_batch[j] = batch[idx[j]]
// 16 threads per row, float4 (b128) loads/stores.
// ---------------------------------------------------------------------------
__global__ __launch_bounds__(256) void gather_kernel(
    const float* __restrict__ x, const float* __restrict__ zp,
    const int* __restrict__ idx, const int* __restrict__ ebatch,
    float* __restrict__ xp, int* __restrict__ nb) {
  const int j  = blockIdx.x * 16 + (threadIdx.x >> 4);
  const int fg = threadIdx.x & 15;
  const int row = idx[j];
  float s = zp[row];
  s = s > 0.0f ? s : 0.0f;
  float4 val = ((const float4*)(x + (size_t)row * F_IN))[fg];
  val.x *= s; val.y *= s; val.z *= s; val.w *= s;
  ((float4*)(xp + (size_t)j * F_IN))[fg] = val;
  if (fg == 0) nb[j] = ebatch[row];
}

// ---------------------------------------------------------------------------
// K4a: partial readout. Each block owns one (graph, slice) pair — 512 blocks
// so the 128 MB x_p re-read can actually saturate the 23.3 TB/s HBM roof
// (64 blocks could not). WMMA with A = ones(16x4): D[0][n] accumulates 4
// x_p rows per instruction; fixed-order LDS cross-wave reduce => determinism.
// ---------------------------------------------------------------------------
__global__ __launch_bounds__(512) void readout_partial_kernel(
    const float* __restrict__ xp, float* __restrict__ partial) {
  const int b = blockIdx.x / RO_SLICES;
  const int s = blockIdx.x % RO_SLICES;
  const int t = threadIdx.x;
  const int wid = t >> 5, lane = t & 31;
  const int n = lane & 15, half = lane >> 4;
  const float* base =
      xp + ((size_t)b * KKEEP + (size_t)s * SLICE_ROWS) * F_IN;

  v2f a; a.x = 1.0f; a.y = 1.0f;  // ones for every (M,K)
  v8f a0 = {}, a1 = {}, a2 = {}, a3 = {};
  for (int r = wid * 4; r < SLICE_ROWS; r += 64) {
    const float* r0 = base + (size_t)(r + half * 2) * F_IN;  // K rows {0,1}/{2,3}
    const float* r1 = r0 + F_IN;
    v2f b0; b0.x = r0[n];      b0.y = r1[n];
    v2f b1; b1.x = r0[16 + n]; b1.y = r1[16 + n];
    v2f b2; b2.x = r0[32 + n]; b2.y = r1[32 + n];
    v2f b3; b3.x = r0[48 + n]; b3.y = r1[48 + n];
    a0 = wmma_f32_4(a, b0, a0);
    a1 = wmma_f32_4(a, b1, a1);
    a2 = wmma_f32_4(a, b2, a2);
    a3 = wmma_f32_4(a, b3, a3);
  }
  __shared__ float part[16 * F_IN];
  if (lane < 16) {  // VGPR0 lanes 0-15 hold D[0][n]
    part[wid * F_IN +  0 + lane] = a0[0];
    part[wid * F_IN + 16 + lane] = a1[0];
    part[wid * F_IN + 32 + lane] = a2[0];
    part[wid * F_IN + 48 + lane] = a3[0];
  }
  __syncthreads();
  if (t < F_IN) {
    float sum = 0.0f;
    for (int w = 0; w < 16; ++w) sum += part[w * F_IN + t];
    partial[((size_t)b * RO_SLICES + s) * F_IN + t] = sum;
  }
}

// K4b: deterministic fixed-order slice reduction -> ro[64,64]
__global__ __launch_bounds__(64) void readout_final_kernel(
    const float* __restrict__ partial, float* __restrict__ ro) {
  const int b = blockIdx.x;
  const int f = threadIdx.x;
  float sum = 0.0f;
#pragma unroll
  for (int s = 0; s < RO_SLICES; ++s)
    sum += partial[((size_t)b * RO_SLICES + s) * F_IN + f];
  ro[b * F_IN + f] = sum;
}

// ---------------------------------------------------------------------------
// K5: perm = -1; perm[idx[j]] = j
// ---------------------------------------------------------------------------
__global__ __launch_bounds__(256) void perm_init_kernel(int* __restrict__ perm) {
  perm[blockIdx.x * 256 + threadIdx.x] = -1;
}
__global__ __launch_bounds__(256) void perm_scatter_kernel(
    const int* __restrict__ idx, int* __restrict__ perm) {
  const int j = blockIdx.x * 256 + threadIdx.x;
  perm[idx[j]] = j;
}

// ---------------------------------------------------------------------------
// K6: filter_adj — reindex both edge endpoints; mask = both kept.
// ---------------------------------------------------------------------------
__global__ __launch_bounds__(256) void filter_kernel(
    const int* __restrict__ edges, const int* __restrict__ perm,
    int* __restrict__ out_edges, float* __restrict__ mask) {
  const int e = blockIdx.x * 256 + threadIdx.x;
  const int r = perm[edges[e]];
  const int c = perm[edges[N_EDGES + e]];
  const bool keep = (r >= 0) && (c >= 0);
  out_edges[e]           = keep ? r : -1;
  out_edges[N_EDGES + e] = keep ? c : -1;
  mask[e] = keep ? 1.0f : 0.0f;
}

// ---------------------------------------------------------------------------
extern "C" void kernel_launch(void* const* d_in, const int* in_sizes, int n_in,
                              void* d_out, int out_size, void* d_ws,
                              size_t ws_size, hipStream_t stream) {
  (void)in_sizes; (void)n_in; (void)out_size; (void)ws_size;
  const float* x      = (const float*)d_in[0];
  const float* w      = (const float*)d_in[1];
  const int*   ebatch = (const int*)d_in[2];
  const int*   up     = (const int*)d_in[3];
  const int*   dn     = (const int*)d_in[4];

  // workspace: zp[N_TOTAL] f32 | idx[N_KEPT] i32 | perm[N_TOTAL] i32
  //            | ro partials [64*8*64] f32                      (~10.1 MB)
  float* zp    = (float*)d_ws;
  int*   idx   = (int*)((char*)d_ws + (size_t)N_TOTAL * 4);
  int*   perm  = (int*)((char*)d_ws + (size_t)N_TOTAL * 4 + (size_t)N_KEPT * 4);
  float* ropar = (float*)((char*)d_ws + (size_t)N_TOTAL * 8 + (size_t)N_KEPT * 4);

  // outputs, concatenated flat in reference return order
  float* out_f   = (float*)d_out;
  float* xp      = out_f;                                   // [N_KEPT, 64] f32
  float* ro      = xp + (size_t)N_KEPT * F_IN;              // [64, 64] f32
  int*   nb      = (int*)(ro + (size_t)N_GRAPHS * F_IN);    // [N_KEPT] i32
  int*   up_new  = nb + (size_t)N_KEPT;                     // [2, N_EDGES] i32
  float* up_mask = (float*)(up_new + (size_t)2 * N_EDGES);  // [N_EDGES] bool->f32
  int*   do_new  = (int*)(up_mask + (size_t)N_EDGES);       // [2, N_EDGES] i32
  float* do_mask = (float*)(do_new + (size_t)2 * N_EDGES);  // [N_EDGES]

  score_kernel<<<N_TOTAL / 16 / 8, 256, 0, stream>>>(x, w, zp);
  topk_kernel<<<N_GRAPHS, 1024, 0, stream>>>(zp, idx);
  gather_kernel<<<N_KEPT / 16, 256, 0, stream>>>(x, zp, idx, ebatch, xp, nb);
  readout_partial_kernel<<<N_GRAPHS * RO_SLICES, 512, 0, stream>>>(xp, ropar);
  readout_final_kernel<<<N_GRAPHS, 64, 0, stream>>>(ropar, ro);
  perm_init_kernel<<<N_TOTAL / 256, 256, 0, stream>>>(perm);
  perm_scatter_kernel<<<N_KEPT / 256, 256, 0, stream>>>(idx, perm);
  filter_kernel<<<N_EDGES / 256, 256, 0, stream>>>(up, perm, up_new, up_mask);
  filter_kernel<<<N_EDGES / 256, 256, 0, stream>>>(dn, perm, do_new, do_mask);
}